// KAConv_6390911336674
// MI455X (gfx1250) — compile-verified
//
#include <hip/hip_runtime.h>

typedef float v2f __attribute__((ext_vector_type(2)));
typedef float v8f __attribute__((ext_vector_type(8)));

#define NB 4
#define NC 16
#define NO 32
#define NK 3
#define NH 56
#define NW 56
#define HW (NH * NW)      // 3136
#define TILES (HW / 16)   // 196 16-pixel tiles per (b,c) plane

// One wave32 per (batch b, 16-pixel tile, 16-output chunk).
// Per (c,i,j) tap:
//   num[16pix x 16o] = Xpow6 * NRcoefs^T   via 2x V_WMMA_F32_16X16X4_F32 (K padded 6->8)
//   den[16pix x 16o] = Xpow4 * DRcoefs^T   via 1x V_WMMA_F32_16X16X4_F32
//   acc += num * rcp(1 + |den|)            elementwise in the shared C/D layout
__global__ __launch_bounds__(256) void kaconv_wmma_kernel(
    const float* __restrict__ x,      // [B, C, H, W]
    const float* __restrict__ nums,   // [O*C*K*K, 6]
    const float* __restrict__ denoms, // [O*C*K*K, 4]
    const float* __restrict__ bias,   // [O]
    float* __restrict__ out)          // [B, O, H, W]
{
  const int lane   = threadIdx.x & 31;
  const int wib    = threadIdx.x >> 5;
  const int wave   = blockIdx.x * 8 + wib;       // 196 blocks * 8 waves = 1568
  const int b      = wave / (TILES * 2);
  const int rem    = wave - b * (TILES * 2);
  const int tile   = rem >> 1;
  const int ochunk = rem & 1;

  const int m  = lane & 15;   // A: pixel row M;  B/C/D: output column N
  const int hi = lane >> 4;   // 0: lanes 0-15 (K=0,1 / M=0..7), 1: lanes 16-31 (K=2,3 / M=8..15)

  const int pix = tile * 16 + m;  // flattened (h,w) of this lane's A-row
  const int h = pix / NW;
  const int w = pix - h * NW;

  v8f acc = {};

  const int obase = ochunk * 16;
  const int orow  = (obase + m) * NC;  // coefficient row prefix for o = obase+m

  for (int c = 0; c < NC; ++c) {
    const float* xc = x + ((size_t)(b * NC + c)) * HW;
    const int rowc = (orow + c) * NK;
#pragma unroll
    for (int i = 0; i < NK; ++i) {
      const int hh    = h + i - 1;
      const int rowci = (rowc + i) * NK;
#pragma unroll
      for (int j = 0; j < NK; ++j) {
        const int ww = w + j - 1;
        const bool inb = (hh >= 0) && (hh < NH) && (ww >= 0) && (ww < NW);
        const float xv = inb ? xc[hh * NW + ww] : 0.0f;  // zero-padded patch value
        const float x2 = xv * xv;
        const float x3 = x2 * xv;
        const float x4 = x2 * x2;
        const float x5 = x3 * x2;

        // A matrices (16x4 f32 layout: lo half K={0,1}, hi half K={2,3})
        v2f aN0 = {}, aN1 = {}, aD = {};
        aN0[0] = hi ? x2 : 1.0f;  aN0[1] = hi ? x3 : xv;   // powers p0..p3
        aN1[0] = hi ? 0.0f : x4;  aN1[1] = hi ? 0.0f : x5; // powers p4,p5 (K=6,7 == 0)
        aD[0]  = hi ? x3 : xv;    aD[1]  = hi ? x4 : x2;   // Q uses x^1..x^4

        // B matrices (4x16 f32, mirror layout): lane = output column,
        // lo half rows K={0,1}, hi half rows K={2,3}. 8-byte aligned float2 loads.
        const int row = rowci + j;
        const float* nr = nums   + (size_t)row * 6;
        const float* dr = denoms + (size_t)row * 4;
        v2f bN0 = *(const v2f*)(nr + (hi ? 2 : 0));
        v2f bN1 = *(const v2f*)(nr + (hi ? 2 : 4)); // hi-half don't-care: A cols 6,7 are 0
        v2f bD  = *(const v2f*)(dr + (hi ? 2 : 0));

        v8f numv = {};
        numv = __builtin_amdgcn_wmma_f32_16x16x4_f32(false, aN0, false, bN0,
                                                     (short)0, numv, false, false);
        numv = __builtin_amdgcn_wmma_f32_16x16x4_f32(false, aN1, false, bN1,
                                                     (short)0, numv, false, false);
        v8f denv = {};
        denv = __builtin_amdgcn_wmma_f32_16x16x4_f32(false, aD, false, bD,
                                                     (short)0, denv, false, false);

        // Elementwise rational combine in the shared 16x16 C/D layout.
#pragma unroll
        for (int k = 0; k < 8; ++k) {
          const float d = 1.0f + __builtin_fabsf(denv[k]);
          acc[k] = __builtin_fmaf(numv[k], __builtin_amdgcn_rcpf(d), acc[k]);
        }
      }
    }
  }

  // C/D layout: lane holds column o = obase+m, rows M = k + 8*hi (pixel index in tile).
  const float bo = bias[obase + m];
  float* op = out + ((size_t)(b * NO + obase + m)) * HW + tile * 16 + hi * 8;
#pragma unroll
  for (int k = 0; k < 8; ++k) op[k] = acc[k] + bo;
}

extern "C" void kernel_launch(void* const* d_in, const int* in_sizes, int n_in,
                              void* d_out, int out_size, void* d_ws, size_t ws_size,
                              hipStream_t stream) {
  const float* x      = (const float*)d_in[0];
  const float* nums   = (const float*)d_in[1];
  const float* denoms = (const float*)d_in[2];
  const float* bias   = (const float*)d_in[3];
  float* out          = (float*)d_out;
  (void)in_sizes; (void)n_in; (void)out_size; (void)d_ws; (void)ws_size;

  dim3 grid(196);   // 196 blocks * 8 waves/block = 1568 waves = 4 * 196 * 2
  dim3 block(256);
  hipLaunchKernelGGL(kaconv_wmma_kernel, grid, block, 0, stream,
                     x, nums, denoms, bias, out);
}